// SimpleRecConv_32341103739244
// MI455X (gfx1250) — compile-verified
//
#include <hip/hip_runtime.h>
#include <hip/hip_bf16.h>

typedef float v2f __attribute__((ext_vector_type(2)));
typedef float v8f __attribute__((ext_vector_type(8)));

#define LDS_STRIDE 132   // 16B-aligned row stride (132*4 = 528 = 16*33)

// ---- ordered-uint encoding of float for atomicMax (handles negatives) ----
__device__ __forceinline__ unsigned fenc(float f) {
    unsigned u = __float_as_uint(f);
    return (u & 0x80000000u) ? ~u : (u | 0x80000000u);
}
__device__ __forceinline__ float fdec(unsigned u) {
    unsigned v = (u & 0x80000000u) ? (u & 0x7fffffffu) : ~u;
    return __uint_as_float(v);
}
#define ENC_NEG_INF 0x007fffffu   // fenc(-inf)

__device__ __forceinline__ float leaky(float x) { return x > 0.f ? x : 0.01f * x; }

// ---- 16x128 (LDS, row stride 132) @ 128x64 (global, strides sk,sn) -> 4x v8f ----
// fp32 WMMA 16x16x4: A lanes0-15 {M=lane,K=kb,kb+1}, lanes16-31 {M=lane-16,K=kb+2,kb+3}
__device__ __forceinline__ void mm_tile(const float* lds_cat,
                                        const float* __restrict__ B,
                                        int sk, int sn, v8f acc[4]) {
    const int lane = threadIdx.x & 31;
    const int hi = lane >> 4;
    const int nl = lane & 15;
#pragma unroll 4
    for (int kk = 0; kk < 32; ++kk) {
        const int kb = kk * 4 + 2 * hi;
        v2f a;
        const float* ap = lds_cat + nl * LDS_STRIDE + kb;
        a[0] = ap[0];
        a[1] = ap[1];
#pragma unroll
        for (int nt = 0; nt < 4; ++nt) {
            const int n = nt * 16 + nl;
            v2f b;
            b[0] = B[kb * sk + n * sn];
            b[1] = B[(kb + 1) * sk + n * sn];
            acc[nt] = __builtin_amdgcn_wmma_f32_16x16x4_f32(
                false, a, false, b, (short)0, acc[nt], false, false);
        }
    }
}

// ---- gather cat tile (16 edges x 128) into LDS: cols 0-63 = h[dst], 64-127 = h[src]
__device__ __forceinline__ void gather_cat(const float* __restrict__ h,
                                           const int* dIdx, const int* sIdx,
                                           float* lds_cat) {
    const int lane = threadIdx.x & 31;
    for (int u = lane; u < 16 * 32; u += 32) {
        const int row = u >> 5;
        const int q = u & 31;         // float4 slot within row
        const int col4 = q * 4;
        const float* p = (q < 16) ? (h + (size_t)dIdx[row] * 64 + col4)
                                  : (h + (size_t)sIdx[row] * 64 + (col4 - 64));
        float4 v = *(const float4*)p;
        *(float4*)(lds_cat + row * LDS_STRIDE + col4) = v;
    }
}

// ================= kernel 0: init workspace =================
__global__ void k_init(unsigned* segmax, float* segsum, float* hN, int N) {
    int i = blockIdx.x * blockDim.x + threadIdx.x;
    if (i < N) { segmax[i] = ENC_NEG_INF; segsum[i] = 0.f; }
    if (i < N * 64) hN[i] = 0.f;
}

// ================= kernel 1: attention scores =================
__global__ void __launch_bounds__(32)
k_att_score(const float* __restrict__ h, const float* __restrict__ Wa_w,
            const float* __restrict__ Wa_b, const float* __restrict__ a0,
            const int* __restrict__ src, const int* __restrict__ dst,
            int E, float* __restrict__ a_sc, unsigned* __restrict__ segmax) {
    __shared__ float catT[16 * LDS_STRIDE];
    __shared__ int dIdx[16], sIdx[16];
    const int lane = threadIdx.x;
    const int e0 = blockIdx.x * 16;
    if (lane < 16) {
        int e = min(e0 + lane, E - 1);
        dIdx[lane] = dst[e];
        sIdx[lane] = src[e];
    }
    __syncthreads();
    gather_cat(h, dIdx, sIdx, catT);
    __syncthreads();

    v8f acc[4] = {};
    mm_tile(catT, Wa_w, /*sk=*/1, /*sn=*/128, acc);   // B[k][n] = Wa_w[n*128+k]

    const int hi = lane >> 4, nl = lane & 15;
    float s[8];
#pragma unroll
    for (int v = 0; v < 8; ++v) s[v] = 0.f;
#pragma unroll
    for (int nt = 0; nt < 4; ++nt) {
        const int n = nt * 16 + nl;
        const float bias = Wa_b[n];
        const float av = a0[n];
#pragma unroll
        for (int v = 0; v < 8; ++v)
            s[v] += leaky(acc[nt][v] + bias) * av;
    }
    // reduce over the 16 lanes of each half (halves hold disjoint row sets)
#pragma unroll
    for (int off = 1; off < 16; off <<= 1)
#pragma unroll
        for (int v = 0; v < 8; ++v)
            s[v] += __shfl_xor(s[v], off, 32);

    if (nl == 0) {
#pragma unroll
        for (int v = 0; v < 8; ++v) {
            const int m = v + 8 * hi;
            const int e = e0 + m;
            if (e < E) {
                a_sc[e] = s[v];
                atomicMax(&segmax[dIdx[m]], fenc(s[v]));
            }
        }
    }
}

// ================= kernel 2: softmax denominator =================
__global__ void k_denom(const float* __restrict__ a_sc, const int* __restrict__ dst,
                        const unsigned* __restrict__ segmax, float* __restrict__ segsum,
                        int E) {
    int e = blockIdx.x * blockDim.x + threadIdx.x;
    if (e < E) {
        int d = dst[e];
        float ex = __expf(a_sc[e] - fdec(segmax[d]));
        atomicAdd(&segsum[d], ex);
    }
}

// ================= kernel 3: gate + message aggregation =================
__global__ void __launch_bounds__(32)
k_aggregate(const float* __restrict__ h, const float* __restrict__ r,
            const float* __restrict__ a_sc, const unsigned* __restrict__ segmax,
            const float* __restrict__ segsum, const int* __restrict__ src,
            const int* __restrict__ dst, const int* __restrict__ etype,
            int E, float* __restrict__ hN) {
    __shared__ float catT[16 * LDS_STRIDE];
    __shared__ int dIdx[16], sIdx[16], et[16];
    __shared__ float attn[16];
    const int lane = threadIdx.x;
    const int e0 = blockIdx.x * 16;
    if (lane < 16) {
        int e = min(e0 + lane, E - 1);
        int d = dst[e];
        dIdx[lane] = d;
        sIdx[lane] = src[e];
        et[lane] = etype[e];
        float ss = segsum[d];
        attn[lane] = (ss > 0.f) ? __expf(a_sc[e] - fdec(segmax[d])) / ss : 0.f;
    }
    __syncthreads();
    gather_cat(h, dIdx, sIdx, catT);
    __syncthreads();

    const int hi = lane >> 4, nl = lane & 15;
    v8f gacc[4] = {};
#pragma unroll
    for (int rel = 0; rel < 4; ++rel) {
        v8f racc[4] = {};
        mm_tile(catT, r + (size_t)rel * 128 * 64, /*sk=*/64, /*sn=*/1, racc);
#pragma unroll
        for (int nt = 0; nt < 4; ++nt)
#pragma unroll
            for (int v = 0; v < 8; ++v) {
                const int m = v + 8 * hi;
                gacc[nt][v] = (et[m] == rel) ? racc[nt][v] : gacc[nt][v];
            }
    }
#pragma unroll
    for (int nt = 0; nt < 4; ++nt) {
        const int n = nt * 16 + nl;
#pragma unroll
        for (int v = 0; v < 8; ++v) {
            const int m = v + 8 * hi;
            const int e = e0 + m;
            const float g = 1.f / (1.f + __expf(-gacc[nt][v]));
            const float msg = catT[m * LDS_STRIDE + 64 + n] * attn[m] * g;
            if (e < E) atomicAdd(&hN[(size_t)dIdx[m] * 64 + n], msg);
        }
    }
}

// ================= kernel 4: node output =================
__global__ void __launch_bounds__(32)
k_output(const float* __restrict__ h, const float* __restrict__ hN,
         const float* __restrict__ lin_w, const float* __restrict__ lin_b,
         int N, float* __restrict__ out) {
    __shared__ float catT[16 * LDS_STRIDE];
    const int lane = threadIdx.x;
    const int i0 = blockIdx.x * 16;
    for (int u = lane; u < 16 * 32; u += 32) {
        const int row = u >> 5;
        const int q = u & 31;
        const int col4 = q * 4;
        const int node = min(i0 + row, N - 1);
        const float* p = (q < 16) ? (h + (size_t)node * 64 + col4)
                                  : (hN + (size_t)node * 64 + (col4 - 64));
        float4 v = *(const float4*)p;
        *(float4*)(catT + row * LDS_STRIDE + col4) = v;
    }
    __syncthreads();

    v8f acc[4] = {};
    mm_tile(catT, lin_w, /*sk=*/1, /*sn=*/128, acc);  // B[k][n] = lin_w[n*128+k]

    const int hi = lane >> 4, nl = lane & 15;
#pragma unroll
    for (int nt = 0; nt < 4; ++nt) {
        const int n = nt * 16 + nl;
        const float bias = lin_b[n];
#pragma unroll
        for (int v = 0; v < 8; ++v) {
            const int m = v + 8 * hi;
            const int node = i0 + m;
            if (node < N) out[(size_t)node * 64 + n] = leaky(acc[nt][v] + bias);
        }
    }
}

extern "C" void kernel_launch(void* const* d_in, const int* in_sizes, int n_in,
                              void* d_out, int out_size, void* d_ws, size_t ws_size,
                              hipStream_t stream) {
    const float* h     = (const float*)d_in[0];
    const float* r     = (const float*)d_in[1];
    const float* a     = (const float*)d_in[2];   // only a[0] (first 64) used
    const float* Wa_w  = (const float*)d_in[3];
    const float* Wa_b  = (const float*)d_in[4];
    const float* lin_w = (const float*)d_in[5];
    const float* lin_b = (const float*)d_in[6];
    const int*   src   = (const int*)d_in[7];
    const int*   dst   = (const int*)d_in[8];
    const int*   etype = (const int*)d_in[9];
    float* out = (float*)d_out;

    const int N = in_sizes[0] / 64;
    const int E = in_sizes[7];

    // workspace layout
    float*    a_sc   = (float*)d_ws;
    unsigned* segmax = (unsigned*)(a_sc + E);
    float*    segsum = (float*)(segmax + N);
    float*    hN     = segsum + N;

    const int initThreads = 256;
    k_init<<<(N * 64 + initThreads - 1) / initThreads, initThreads, 0, stream>>>(
        segmax, segsum, hN, N);

    const int etiles = (E + 15) / 16;
    k_att_score<<<etiles, 32, 0, stream>>>(h, Wa_w, Wa_b, a, src, dst, E, a_sc, segmax);

    k_denom<<<(E + 255) / 256, 256, 0, stream>>>(a_sc, dst, segmax, segsum, E);

    k_aggregate<<<etiles, 32, 0, stream>>>(h, r, a_sc, segmax, segsum, src, dst,
                                           etype, E, hN);

    const int ntiles = (N + 15) / 16;
    k_output<<<ntiles, 32, 0, stream>>>(h, hN, lin_w, lin_b, N, out);
}